// WHTConv2D_84679575207946
// MI455X (gfx1250) — compile-verified
//
#include <hip/hip_runtime.h>
#include <hip/hip_bf16.h>
#include <math.h>

typedef float v2f __attribute__((ext_vector_type(2)));
typedef float v8f __attribute__((ext_vector_type(8)));

#define BATCH 8
#define CHAN  64
#define HH    256
#define WW    256
#define HWSZ  65536   // H*W

// ---------------------------------------------------------------------------
// FWHT along W: one wave (32 lanes) per 256-element row.
// lane holds elements {lane + 32*j : j=0..7}. Cross-lane stages h=1,2,4,8,16
// via shfl_xor; intra-lane stages h=32,64,128 on the j bits.
// Optionally scales and adds residual (final inverse pass).
// ---------------------------------------------------------------------------
__global__ __launch_bounds__(256) void k_fwht_w(const float* src, float* dst,
                                                const float* resid, float scale) {
    const int lane = threadIdx.x & 31;
    const int wave = threadIdx.x >> 5;
    const size_t row  = (size_t)blockIdx.x * 8 + wave;   // B*C*H rows total
    const size_t base = row * WW;

    float v[8];
#pragma unroll
    for (int j = 0; j < 8; ++j) v[j] = src[base + lane + 32 * j];

    // cross-lane butterfly stages (address bits 0..4)
#pragma unroll
    for (int m = 1; m < 32; m <<= 1) {
#pragma unroll
        for (int j = 0; j < 8; ++j) {
            float o = __shfl_xor(v[j], m, 32);
            v[j] = (lane & m) ? (o - v[j]) : (v[j] + o);
        }
    }
    // intra-lane butterfly stages (address bits 5..7 == j bits 0..2)
#pragma unroll
    for (int hb = 1; hb < 8; hb <<= 1) {
#pragma unroll
        for (int j = 0; j < 8; ++j) {
            if (!(j & hb)) {
                float a = v[j], b = v[j | hb];
                v[j] = a + b;
                v[j | hb] = a - b;
            }
        }
    }

    if (resid) {
#pragma unroll
        for (int j = 0; j < 8; ++j) {
            size_t idx = base + lane + 32 * j;
            dst[idx] = v[j] * scale + resid[idx];
        }
    } else {
#pragma unroll
        for (int j = 0; j < 8; ++j) dst[base + lane + 32 * j] = v[j];
    }
}

// ---------------------------------------------------------------------------
// FWHT along H: block owns a 256(H) x 32(W) tile of one (b,c) slab in LDS.
// 8 butterfly stages with barriers. In-place safe (tile fully loaded first).
// ---------------------------------------------------------------------------
__global__ __launch_bounds__(256) void k_fwht_h(const float* src, float* dst) {
    __shared__ float tile[HH * 32];   // 32 KB
    const int tid  = threadIdx.x;
    const int wt   = blockIdx.x & 7;       // 8 w-tiles of 32
    const int slab = blockIdx.x >> 3;      // b*C + c, 0..511
    const size_t sbase = (size_t)slab * HWSZ + (size_t)wt * 32;

    const int w4 = tid & 7;       // which float4 in the 32-wide tile
    const int hr = tid >> 3;      // 0..31
#pragma unroll
    for (int hb = 0; hb < 8; ++hb) {
        int h = hb * 32 + hr;
        float4 d = *(const float4*)(src + sbase + (size_t)h * WW + w4 * 4);
        *(float4*)(tile + h * 32 + w4 * 4) = d;
    }
    __syncthreads();

    for (int st = 1; st < HH; st <<= 1) {
        for (int p = tid; p < 128 * 32; p += 256) {
            int w = p & 31;
            int q = p >> 5;                               // 0..127
            int i = ((q & ~(st - 1)) << 1) | (q & (st - 1)); // bit-st clear
            float a = tile[i * 32 + w];
            float b = tile[(i + st) * 32 + w];
            tile[i * 32 + w]        = a + b;
            tile[(i + st) * 32 + w] = a - b;
        }
        __syncthreads();
    }

#pragma unroll
    for (int hb = 0; hb < 8; ++hb) {
        int h = hb * 32 + hr;
        *(float4*)(dst + sbase + (size_t)h * WW + w4 * 4) =
            *(float4*)(tile + h * 32 + w4 * 4);
    }
}

// ---------------------------------------------------------------------------
// Channel mix: out[b,o,pix] = softthresh( sum_c Wq[o,c] * f[b,c,pix], T[pix] )
// Wq built in LDS in the WMMA f32 16x16x4 A-operand pair layout:
//   sA[(2k+g)*64 + row] = (Wq[row][4k+2g], Wq[row][4k+2g+1])
// Each wave: 16-pixel strip, 4 accumulators (o-tiles of 16), 16 K-steps,
// 64 x v_wmma_f32_16x16x4_f32 per wave. In-place safe.
// ---------------------------------------------------------------------------
__global__ __launch_bounds__(256) void k_mix(const float* src, float* dst,
                                             const float* wr, const float* wi,
                                             const float* wj, const float* wk,
                                             const float* T) {
    __shared__ v2f sA[32 * 64];   // 16 KB: A-operand pairs of Wq
    const int tid = threadIdx.x;

    {   // build Wq (Hamilton product block matrix) directly in A layout
        float* f = (float*)sA;
        const float* wsrc[4] = {wr, wi, wj, wk};
        const int   midx[16] = {0,1,2,3, 1,0,3,2, 2,3,0,1, 3,2,1,0};
        const float msgn[16] = {1.f,-1.f,-1.f,-1.f,
                                1.f, 1.f,-1.f, 1.f,
                                1.f, 1.f, 1.f,-1.f,
                                1.f,-1.f, 1.f, 1.f};
        for (int idx = tid; idx < 64 * 64; idx += 256) {
            int o  = idx >> 6, c = idx & 63;
            int bi = o >> 4, oi = o & 15, bj = c >> 4, cj = c & 15;
            int sel = bi * 4 + bj;
            float val = msgn[sel] * wsrc[midx[sel]][oi * 16 + cj];
            int pr = c >> 1;
            f[(pr * 64 + o) * 2 + (c & 1)] = val;
        }
    }
    __syncthreads();

    const int lane = tid & 31;
    const int wave = tid >> 5;
    const int g = lane >> 4;       // lane half (selects K pair within step)
    const int n = lane & 15;       // N (pixel) / M (row) index within 16
    const int bb = blockIdx.x >> 9;                       // batch
    const int pixBase = (((blockIdx.x & 511) * 8) + wave) * 16;
    const int pix = pixBase + n;

    const float thr = T[pix];
    const float* Fb = src + (size_t)bb * CHAN * HWSZ;
    float*       Ob = dst + (size_t)bb * CHAN * HWSZ;

    v8f acc[4] = {};
#pragma unroll
    for (int k = 0; k < 16; ++k) {
        const int c0 = 4 * k + 2 * g;
        v2f bvec;
        bvec.x = Fb[(size_t)c0 * HWSZ + pix];        // B: K=4k+2g,  N=n
        bvec.y = Fb[(size_t)(c0 + 1) * HWSZ + pix];  // B: K=4k+2g+1,N=n
#pragma unroll
        for (int ot = 0; ot < 4; ++ot) {
            v2f avec = sA[(2 * k + g) * 64 + ot * 16 + n];  // A: M=ot*16+n
            acc[ot] = __builtin_amdgcn_wmma_f32_16x16x4_f32(
                false, avec, false, bvec, (short)0, acc[ot], false, false);
        }
    }

    // D layout: VGPR r, lane-half g -> M = r + 8*g ; N = n. Soft-threshold.
#pragma unroll
    for (int ot = 0; ot < 4; ++ot) {
#pragma unroll
        for (int r = 0; r < 8; ++r) {
            int o = ot * 16 + 8 * g + r;
            float v = acc[ot][r];
            float mag = fabsf(v) - thr;
            mag = mag > 0.f ? mag : 0.f;
            v = copysignf(mag, v);
            Ob[(size_t)o * HWSZ + pix] = v;
        }
    }
}

// ---------------------------------------------------------------------------
extern "C" void kernel_launch(void* const* d_in, const int* in_sizes, int n_in,
                              void* d_out, int out_size, void* d_ws, size_t ws_size,
                              hipStream_t stream) {
    const float* x  = (const float*)d_in[0];
    const float* wr = (const float*)d_in[1];
    const float* wi = (const float*)d_in[2];
    const float* wj = (const float*)d_in[3];
    const float* wk = (const float*)d_in[4];
    const float* T  = (const float*)d_in[5];
    float* out = (float*)d_out;

    const int rowsBlocks = (BATCH * CHAN * HH) / 8;   // 16384 (8 rows/block)
    const int colBlocks  = (BATCH * CHAN) * 8;        // 4096  (8 w-tiles/slab)
    const int mixBlocks  = BATCH * (HWSZ / (8 * 16)); // 4096  (8 strips/block)

    // 1) FWHT along W :  x -> out
    k_fwht_w<<<rowsBlocks, 256, 0, stream>>>(x, out, nullptr, 1.0f);
    // 2) FWHT along H :  out -> out (in-place, tile-local)
    k_fwht_h<<<colBlocks, 256, 0, stream>>>(out, out);
    // 3) channel mix + soft-threshold (WMMA f32) : out -> out (in-place)
    k_mix<<<mixBlocks, 256, 0, stream>>>(out, out, wr, wi, wj, wk, T);
    // 4) inverse FWHT along H (unnormalized) : out -> out
    k_fwht_h<<<colBlocks, 256, 0, stream>>>(out, out);
    // 5) inverse FWHT along W + 1/65536 scale + residual : out -> out
    k_fwht_w<<<rowsBlocks, 256, 0, stream>>>(out, out, x, 1.0f / 65536.0f);
}